// HGT_89146341196549
// MI455X (gfx1250) — compile-verified
//
#include <hip/hip_runtime.h>
#include <hip/hip_bf16.h>

// ---------------------------------------------------------------------------
// HGT forward (2 shared-weight layers) for MI455X / gfx1250.
//  - Typed-linear projections: native f32 WMMA (v_wmma_f32_16x16x4_f32).
//    Weights are repacked once into a WMMA-native layout so every B operand
//    is a single aligned global_load_b64; the strip kernel is explicitly
//    double-buffered so loads for k-step s+1 are in flight during the 8
//    WMMAs of step s (partial s_wait_loadcnt, no full stalls).
//  - Edge relation transforms: VALU FMA reading relation matrices from LDS
//    (32KB staged per block, bank-conflict-free padding).
//  - Edge softmax / aggregation: ordered-uint atomicMax + f32 atomicAdd,
//    matching segment_max / segment_sum semantics.
// ---------------------------------------------------------------------------

namespace {
constexpr int kH      = 8;
constexpr int kD      = 16;
constexpr int kIn     = 128;           // kH * kD
constexpr int kET     = 4;
constexpr int kNT     = 3;
constexpr int kNDrug  = 10000;
constexpr int kNDis   = 10000;
constexpr int kN      = 50000;         // drug + disease + protein
constexpr int kE      = 400000;
constexpr int kLayers = 2;
constexpr int kMTiles = kN / 16;       // 3125 (all type boundaries are /16)
constexpr int kNTiles = kIn / 16;      // 8
constexpr int kWElems = kIn * kIn;     // 16384 floats per type-matrix
// LDS slot stride for one 16x16 relation matrix: 258 floats so that the
// per-head stride (4*258 = 1032 = 8 mod 64 banks) spreads 8 heads over 8
// distinct LDS banks.
constexpr int kRelPad = 258;
constexpr int kRelElems = kH * kET * kD * kD;   // 8192 floats
}

typedef __attribute__((ext_vector_type(2)))  float  v2f;
typedef __attribute__((ext_vector_type(8)))  float  v8f;

// ----------------------------- small helpers ------------------------------

__device__ __forceinline__ int node_type(int row) {
  return row < kNDrug ? 0 : (row < kNDrug + kNDis ? 1 : 2);
}

// monotone float <-> uint mapping so unsigned atomicMax == float max
__device__ __forceinline__ unsigned f2ord(float f) {
  unsigned u = __float_as_uint(f);
  return (u & 0x80000000u) ? ~u : (u | 0x80000000u);
}
__device__ __forceinline__ float ord2f(unsigned u) {
  unsigned v = (u & 0x80000000u) ? (u & 0x7fffffffu) : ~u;
  return __uint_as_float(v);
}

// ---------------------------------------------------------------------------
// Weight repack: row-major W[ka][n] (128x128) -> Wp[((step*2+hi)*128+n)*2+sub]
// where ka = step*4 + hi*2 + sub.  Each lane's WMMA B operand {W[ka][n],
// W[ka+1][n]} becomes one contiguous, 8B-aligned b64 load.
// ---------------------------------------------------------------------------
__global__ void hgt_pack_w(const float* __restrict__ W,
                           float* __restrict__ Wp) {
  const int i = blockIdx.x * blockDim.x + threadIdx.x;   // over kNT * 128*128
  if (i >= kNT * kWElems) return;
  const int m     = i >> 14;            // type-matrix index
  const int inner = i & (kWElems - 1);
  const int ka = inner >> 7, n = inner & 127;
  const int step = ka >> 2, hi = (ka >> 1) & 1, sub = ka & 1;
  Wp[(size_t)m * kWElems + ((((step << 1) + hi) << 7) + n) * 2 + sub] = W[i];
}

// ---------------------------------------------------------------------------
// One wave: 16-row strip [m0, m0+16) x ALL 128 columns of  X @ W  (K = 128),
// v_wmma_f32_16x16x4_f32, 32 k-steps x 8 column tiles, double-buffered.
// A (16x4 f32): lanes 0-15 row=lane, K=kk,kk+1; lanes 16-31 row=lane-16,
// K=kk+2,kk+3.  B (4x16): col = lane&15, same K split (from packed Wp).
// C/D: VGPR r -> row r (lanes 0-15) / r+8 (lanes 16-31), col = lane&15.
// ---------------------------------------------------------------------------
__device__ __forceinline__ void gemm_strip_128(const float* __restrict__ X,
                                               const float* __restrict__ Wp,
                                               int m0, int lane,
                                               v8f acc[kNTiles]) {
  const int lm = lane & 15;
  const int hi = lane >> 4;
  #pragma unroll
  for (int nt = 0; nt < kNTiles; ++nt)
    acc[nt] = (v8f){0.f, 0.f, 0.f, 0.f, 0.f, 0.f, 0.f, 0.f};
  const float* xrow  = X + (size_t)(m0 + lm) * kIn + 2 * hi;
  const float* wbase = Wp + hi * 256 + lm * 2;

  v2f a_c = *(const v2f*)(xrow);
  v2f b_c[kNTiles];
  #pragma unroll
  for (int nt = 0; nt < kNTiles; ++nt)
    b_c[nt] = *(const v2f*)(wbase + nt * 32);

  #pragma unroll
  for (int step = 0; step < 32; ++step) {
    v2f a_n = a_c;
    v2f b_n[kNTiles];
    #pragma unroll
    for (int nt = 0; nt < kNTiles; ++nt) b_n[nt] = b_c[nt];
    if (step < 31) {
      a_n = *(const v2f*)(xrow + (step + 1) * 4);
      #pragma unroll
      for (int nt = 0; nt < kNTiles; ++nt)
        b_n[nt] = *(const v2f*)(wbase + (step + 1) * 512 + nt * 32);
    }
    #pragma unroll
    for (int nt = 0; nt < kNTiles; ++nt)
      acc[nt] = __builtin_amdgcn_wmma_f32_16x16x4_f32(false, a_c, false,
                                                      b_c[nt], (short)0,
                                                      acc[nt], false, false);
    a_c = a_n;
    #pragma unroll
    for (int nt = 0; nt < kNTiles; ++nt) b_c[nt] = b_n[nt];
  }
}

// ------------------------------- kernels ----------------------------------

__global__ void hgt_init_layer(unsigned* __restrict__ amax,
                               float* __restrict__ esum,
                               float* __restrict__ hagg) {
  const int i = blockIdx.x * blockDim.x + threadIdx.x;
  if (i < kN * kH) {
    amax[i] = f2ord(-__builtin_huge_valf());
    esum[i] = 0.f;
  }
  if (i < kN * kIn) hagg[i] = 0.f;
}

// K/Q/V typed projections: wave w handles matrix (w / kMTiles) and
// row strip (w % kMTiles) * 16, producing all 128 output columns.
// Pointer selection via register if/else keeps loads on GLOBAL_LOAD.
__global__ void hgt_gemm_kqv(const float* __restrict__ X,
                             const float* __restrict__ Wkp,
                             const float* __restrict__ Wqp,
                             const float* __restrict__ Wvp,
                             float* __restrict__ Ko,
                             float* __restrict__ Qo,
                             float* __restrict__ Vo) {
  const int lane = threadIdx.x & 31;
  const int gw   = blockIdx.x * (blockDim.x >> 5) + (threadIdx.x >> 5);
  if (gw >= 3 * kMTiles) return;
  const int which = gw / kMTiles;           // 0=K, 1=Q, 2=V (wave-uniform)
  const int m0    = (gw % kMTiles) * 16;
  const size_t wofs = (size_t)node_type(m0) * kWElems;

  const float* W;
  float*       O;
  if (which == 0)      { W = Wkp; O = Ko; }
  else if (which == 1) { W = Wqp; O = Qo; }
  else                 { W = Wvp; O = Vo; }
  W += wofs;

  v8f acc[kNTiles];
  gemm_strip_128(X, W, m0, lane, acc);

  const int lm = lane & 15;
  const int rb = (lane >> 4) * 8;
  #pragma unroll
  for (int nt = 0; nt < kNTiles; ++nt) {
    #pragma unroll
    for (int r = 0; r < 8; ++r)
      O[(size_t)(m0 + rb + r) * kIn + nt * 16 + lm] = acc[nt][r];
  }
}

// a[e,h] = (k[src]@rel_att[h,et]) . q[dst] * pri[h,et] / sqrt(D); seg-max(dst).
__global__ void hgt_edge_logits(const float* __restrict__ K,
                                const float* __restrict__ Q,
                                const int* __restrict__ src,
                                const int* __restrict__ dst,
                                const int* __restrict__ etype,
                                const float* __restrict__ rel_att,
                                const float* __restrict__ pri,
                                float* __restrict__ abuf,
                                unsigned* __restrict__ amax) {
  __shared__ float Rl[kH * kET * kRelPad];     // 33 KB, padded slots
  for (int i = threadIdx.x; i < kRelElems; i += blockDim.x)
    Rl[(i >> 8) * kRelPad + (i & 255)] = rel_att[i];
  __syncthreads();

  const int tid = blockIdx.x * blockDim.x + threadIdx.x;
  if (tid >= kE * kH) return;
  const int e = tid >> 3, h = tid & 7;
  const int s = src[e], d = dst[e], t = etype[e];
  const float* ke = K + (size_t)s * kIn + h * kD;
  const float* qe = Q + (size_t)d * kIn + h * kD;
  const float* R  = &Rl[(h * kET + t) * kRelPad];
  float kv[kD], qv[kD];
  #pragma unroll
  for (int i = 0; i < kD; ++i) { kv[i] = ke[i]; qv[i] = qe[i]; }
  float acc = 0.f;
  #pragma unroll
  for (int o = 0; o < kD; ++o) {
    float kw = 0.f;
    #pragma unroll
    for (int i = 0; i < kD; ++i) kw = fmaf(kv[i], R[i * kD + o], kw);
    acc = fmaf(kw, qv[o], acc);
  }
  acc *= pri[h * kET + t] * 0.25f;               // 1/sqrt(16)
  abuf[(size_t)e * kH + h] = acc;
  atomicMax(&amax[(size_t)d * kH + h], f2ord(acc));
}

// ea = exp(a - amax[dst]); seg-sum(dst).
__global__ void hgt_edge_expsum(const int* __restrict__ dst,
                                const unsigned* __restrict__ amax,
                                float* __restrict__ abuf,
                                float* __restrict__ esum) {
  const int tid = blockIdx.x * blockDim.x + threadIdx.x;
  if (tid >= kE * kH) return;
  const int e = tid >> 3, h = tid & 7;
  const int d = dst[e];
  const float m  = ord2f(amax[(size_t)d * kH + h]);
  const float ea = __expf(abuf[(size_t)e * kH + h] - m);
  abuf[(size_t)e * kH + h] = ea;
  atomicAdd(&esum[(size_t)d * kH + h], ea);
}

// hagg[dst] += attn * (v[src] @ rel_msg[h,et])
__global__ void hgt_edge_message(const float* __restrict__ V,
                                 const int* __restrict__ src,
                                 const int* __restrict__ dst,
                                 const int* __restrict__ etype,
                                 const float* __restrict__ rel_msg,
                                 const float* __restrict__ abuf,
                                 const float* __restrict__ esum,
                                 float* __restrict__ hagg) {
  __shared__ float Rl[kH * kET * kRelPad];
  for (int i = threadIdx.x; i < kRelElems; i += blockDim.x)
    Rl[(i >> 8) * kRelPad + (i & 255)] = rel_msg[i];
  __syncthreads();

  const int tid = blockIdx.x * blockDim.x + threadIdx.x;
  if (tid >= kE * kH) return;
  const int e = tid >> 3, h = tid & 7;
  const int s = src[e], d = dst[e], t = etype[e];
  const float attn = abuf[(size_t)e * kH + h] / esum[(size_t)d * kH + h];
  const float* ve = V + (size_t)s * kIn + h * kD;
  const float* R  = &Rl[(h * kET + t) * kRelPad];
  float vv[kD];
  #pragma unroll
  for (int i = 0; i < kD; ++i) vv[i] = ve[i];
  float* outp = hagg + (size_t)d * kIn + h * kD;
  #pragma unroll
  for (int o = 0; o < kD; ++o) {
    float mw = 0.f;
    #pragma unroll
    for (int i = 0; i < kD; ++i) mw = fmaf(vv[i], R[i * kD + o], mw);
    atomicAdd(&outp[o], attn * mw);
  }
}

// out = (hagg @ Wa[type]) * sigmoid(skip[type]) + x * (1 - sigmoid(skip[type]))
__global__ void hgt_gemm_out(const float* __restrict__ Hin,
                             const float* __restrict__ Wap,
                             const float* __restrict__ Xin,
                             const float* __restrict__ skip,
                             float* __restrict__ Out) {
  const int lane = threadIdx.x & 31;
  const int gw   = blockIdx.x * (blockDim.x >> 5) + (threadIdx.x >> 5);
  if (gw >= kMTiles) return;
  const int m0 = gw * 16;
  const int t  = node_type(m0);
  const float alpha = 1.f / (1.f + __expf(-skip[t]));

  v8f acc[kNTiles];
  gemm_strip_128(Hin, Wap + (size_t)t * kWElems, m0, lane, acc);

  const int lm = lane & 15;
  const int rb = (lane >> 4) * 8;
  #pragma unroll
  for (int nt = 0; nt < kNTiles; ++nt) {
    #pragma unroll
    for (int r = 0; r < 8; ++r) {
      const size_t idx = (size_t)(m0 + rb + r) * kIn + nt * 16 + lm;
      Out[idx] = acc[nt][r] * alpha + Xin[idx] * (1.f - alpha);
    }
  }
}

// ------------------------------ launcher ----------------------------------

extern "C" void kernel_launch(void* const* d_in, const int* in_sizes, int n_in,
                              void* d_out, int out_size, void* d_ws, size_t ws_size,
                              hipStream_t stream) {
  (void)in_sizes; (void)n_in; (void)out_size; (void)ws_size;
  const float* drug    = (const float*)d_in[0];
  const float* dis     = (const float*)d_in[1];
  const float* prot    = (const float*)d_in[2];
  const int*   src     = (const int*)d_in[3];
  const int*   dst     = (const int*)d_in[4];
  const int*   etype   = (const int*)d_in[5];
  const float* Wk      = (const float*)d_in[6];
  const float* Wq      = (const float*)d_in[7];
  const float* Wv      = (const float*)d_in[8];
  const float* Wa      = (const float*)d_in[9];
  const float* rel_att = (const float*)d_in[10];
  const float* rel_msg = (const float*)d_in[11];
  const float* pri     = (const float*)d_in[12];
  const float* skip    = (const float*)d_in[13];
  float* out = (float*)d_out;

  // workspace partition (256B aligned)
  char* ws = (char*)d_ws;
  size_t off = 0;
  auto take = [&](size_t bytes) -> char* {
    char* p = ws + off;
    off = (off + bytes + 255) & ~(size_t)255;
    return p;
  };
  float*    x0   = (float*)take((size_t)kN * kIn * sizeof(float));
  float*    x1   = (float*)take((size_t)kN * kIn * sizeof(float));
  float*    Kb   = (float*)take((size_t)kN * kIn * sizeof(float));
  float*    Qb   = (float*)take((size_t)kN * kIn * sizeof(float));
  float*    Vb   = (float*)take((size_t)kN * kIn * sizeof(float));
  float*    hagg = (float*)take((size_t)kN * kIn * sizeof(float));
  float*    abuf = (float*)take((size_t)kE * kH * sizeof(float));
  unsigned* amax = (unsigned*)take((size_t)kN * kH * sizeof(unsigned));
  float*    esum = (float*)take((size_t)kN * kH * sizeof(float));
  float*    Wkp  = (float*)take((size_t)kNT * kWElems * sizeof(float));
  float*    Wqp  = (float*)take((size_t)kNT * kWElems * sizeof(float));
  float*    Wvp  = (float*)take((size_t)kNT * kWElems * sizeof(float));
  float*    Wap  = (float*)take((size_t)kNT * kWElems * sizeof(float));

  // x = concat(drug, disease, protein)  (graph-capture-safe d2d copies)
  hipMemcpyAsync(x0, drug, (size_t)kNDrug * kIn * sizeof(float),
                 hipMemcpyDeviceToDevice, stream);
  hipMemcpyAsync(x0 + (size_t)kNDrug * kIn, dis,
                 (size_t)kNDis * kIn * sizeof(float),
                 hipMemcpyDeviceToDevice, stream);
  hipMemcpyAsync(x0 + (size_t)(kNDrug + kNDis) * kIn, prot,
                 (size_t)(kN - kNDrug - kNDis) * kIn * sizeof(float),
                 hipMemcpyDeviceToDevice, stream);

  // one-time weight repack into WMMA-native b64 layout
  const dim3 pack_grid((kNT * kWElems + 255) / 256);
  hgt_pack_w<<<pack_grid, 256, 0, stream>>>(Wk, Wkp);
  hgt_pack_w<<<pack_grid, 256, 0, stream>>>(Wq, Wqp);
  hgt_pack_w<<<pack_grid, 256, 0, stream>>>(Wv, Wvp);
  hgt_pack_w<<<pack_grid, 256, 0, stream>>>(Wa, Wap);

  const dim3 gemm_blk(128);                        // 4 waves/block
  const dim3 kqv_grid((3 * kMTiles + 3) / 4);      // 9375 waves
  const dim3 out_grid((kMTiles + 3) / 4);          // 3125 waves
  const int  eth = kE * kH;
  const dim3 edge_blk(256);
  const dim3 edge_grid((eth + 255) / 256);
  const dim3 init_grid(((size_t)kN * kIn + 255) / 256);

  for (int layer = 0; layer < kLayers; ++layer) {
    const float* xin  = (layer == 0) ? x0 : x1;
    float*       xout = (layer == kLayers - 1) ? out : x1;

    hgt_init_layer<<<init_grid, 256, 0, stream>>>(amax, esum, hagg);
    hgt_gemm_kqv<<<kqv_grid, gemm_blk, 0, stream>>>(xin, Wkp, Wqp, Wvp,
                                                    Kb, Qb, Vb);
    hgt_edge_logits<<<edge_grid, edge_blk, 0, stream>>>(Kb, Qb, src, dst, etype,
                                                        rel_att, pri, abuf, amax);
    hgt_edge_expsum<<<edge_grid, edge_blk, 0, stream>>>(dst, amax, abuf, esum);
    hgt_edge_message<<<edge_grid, edge_blk, 0, stream>>>(Vb, src, dst, etype,
                                                         rel_msg, abuf, esum, hagg);
    hgt_gemm_out<<<out_grid, gemm_blk, 0, stream>>>(hagg, Wap, xin, skip, xout);
  }
}